// SGVAE_12146167513746
// MI455X (gfx1250) — compile-verified
//
#include <hip/hip_runtime.h>
#include <hip/hip_bf16.h>

// ---------------------------------------------------------------------------
// GraphProp (gather -> msg Linear -> scatter-sum -> GRUCell) x ROUNDS on
// MI455X (gfx1250, wave32). All GEMMs run on V_WMMA_F32_16X16X4_F32 (exact
// fp32, like the reference). The edge kernel stages its weight slab with the
// Tensor Data Mover (TENSOR_LOAD_TO_LDS + s_wait_tensorcnt), then permutes it
// into a K-pair-interleaved LDS layout so every B fragment is one
// ds_load_2addr_b64 straight into the WMMA source register pair (no movs).
//
// Sizes: N=50000, E=800000, D=64, EDGE_DIM=4, ROUNDS=2.
// E/16 = 50000 tiles, N/16 = 3125 tiles -> no remainders; EXEC is all-ones
// around every WMMA (ISA requirement).
// ---------------------------------------------------------------------------

#define N_NODES  50000
#define N_EDGES  800000
#define D        64        // NODE_DIM == MSG_DIM
#define EDIM     4
#define IN_DIM   (2*D + EDIM)   // 132
#define H3       (3*D)          // 192
#define ROUNDS   2

typedef float v2f __attribute__((ext_vector_type(2)));
typedef float v8f __attribute__((ext_vector_type(8)));
typedef unsigned int v4u __attribute__((ext_vector_type(4)));
typedef int v4i __attribute__((ext_vector_type(4)));
typedef int v8i __attribute__((ext_vector_type(8)));

static __device__ __forceinline__ v8f wmma4(v2f a, v2f b, v8f c) {
  // D(16x16,f32) = A(16x4,f32) * B(4x16,f32) + C
  return __builtin_amdgcn_wmma_f32_16x16x4_f32(
      /*neg_a=*/false, a, /*neg_b=*/false, b,
      /*c_mod=*/(short)0, c, /*reuse_a=*/false, /*reuse_b=*/false);
}

static __device__ __forceinline__ v8f splat8(float v) {
  v8f r;
  #pragma unroll
  for (int i = 0; i < 8; ++i) r[i] = v;
  return r;
}

#define LOG2E 1.4426950408889634f
static __device__ __forceinline__ float fast_sigmoid(float x) {
  float e = __builtin_amdgcn_exp2f(-x * LOG2E);
  return __builtin_amdgcn_rcpf(1.0f + e);
}
static __device__ __forceinline__ float fast_tanh(float x) {
  float e = __builtin_amdgcn_exp2f(2.0f * LOG2E * x);
  return 1.0f - 2.0f * __builtin_amdgcn_rcpf(e + 1.0f);
}

// ---------------------------------------------------------------------------
// Tensor Data Mover: DMA a flat f32 slab from global memory into LDS at byte
// offset lds_off. Issued per-wave (EXEC ignored); tracked by TENSORcnt.
// ---------------------------------------------------------------------------
static __device__ __forceinline__ void tdm_load_1d_to_lds(
    const void* gptr, unsigned lds_off, unsigned nelems)
{
  unsigned long long ga = (unsigned long long)(size_t)gptr;
  v4u g0;
  g0[0] = 1u;                                          // count=1, user D#
  g0[1] = lds_off;                                     // lds_addr (bytes)
  g0[2] = (unsigned)(ga & 0xFFFFFFFFu);                // global_addr lo
  g0[3] = (unsigned)((ga >> 32) & 0x1FFFFFFu)          // global_addr hi
          | (2u << 30);                                // type=2 ("image")
  v8i g1;
  g1[0] = (int)(2u << 16);                             // data_size=4B, mask=0
  g1[1] = (int)((nelems & 0xFFFFu) << 16);             // tensor_dim0 lo16
  g1[2] = (int)(((nelems >> 16) & 0xFFFFu) | (1u << 16)); // td0 hi | td1=1
  g1[3] = (int)((nelems & 0xFFFFu) << 16);             // td1 hi=0 | tile_dim0
  g1[4] = 1;                                           // tile_dim1=1
  g1[5] = (int)nelems;                                 // dim0_stride lo32
  g1[6] = 0;
  g1[7] = 0;
  v4i z4 = {0, 0, 0, 0};
#if __clang_major__ >= 23
  v8i z8 = {0, 0, 0, 0, 0, 0, 0, 0};
  __builtin_amdgcn_tensor_load_to_lds(g0, g1, z4, z4, z8, 0);
#else
  __builtin_amdgcn_tensor_load_to_lds(g0, g1, z4, z4, 0);
#endif
}

// --------------------------------------------------------------------------
// zero-fill for the scatter accumulator (ws is poisoned by the harness)
// --------------------------------------------------------------------------
__global__ void gp_fill0(float* __restrict__ p, int n) {
  for (int i = blockIdx.x * blockDim.x + threadIdx.x; i < n;
       i += gridDim.x * blockDim.x)
    p[i] = 0.0f;
}

// --------------------------------------------------------------------------
// Edge kernel: per 16-edge tile (one wave):
//   msg[16,64] = hv[src]•Ws + hv[dst]•Wd + he•We + b    (33 WMMAs / out-tile)
//   atomicAdd into a[dst[e], :]                          (L2 f32 atomics)
// W_msg staged by TDM, then interleaved: sW[((k>>1)*64 + n)*2 + (k&1)].
// --------------------------------------------------------------------------
__global__ void __launch_bounds__(256)
gp_edge_msg(const float* __restrict__ hv,     // [N, 64]
            const float* __restrict__ he,     // [E, 4]
            const float* __restrict__ Wmsg,   // [132, 64] (round slice)
            const float* __restrict__ bmsg,   // [64]
            const int*   __restrict__ src,    // [E]
            const int*   __restrict__ dst,    // [E]
            float*       __restrict__ acc_a)  // [N, 64] scatter target
{
  __shared__ float stage[IN_DIM * D];   // 33 KB natural layout (TDM target)
  __shared__ float sW[IN_DIM * D];      // 33 KB K-pair interleaved

  if (threadIdx.x < 32) {               // wave 0 drives the DMA
    tdm_load_1d_to_lds(Wmsg, (unsigned)(size_t)&stage[0], IN_DIM * D);
    __builtin_amdgcn_s_wait_tensorcnt(0);
  }
  asm volatile("" ::: "memory");        // LDS 'stage' now holds the weights
  __syncthreads();

  // one-time permute: natural [k][n] -> interleaved [(k>>1)][n][k&1]
  for (int i = threadIdx.x; i < IN_DIM * D; i += blockDim.x) {
    int k = i >> 6, n = i & 63;
    sW[(((k >> 1) * D) + n) * 2 + (k & 1)] = stage[i];
  }
  __syncthreads();

  const int lane = threadIdx.x & 31;
  const int row  = lane & 15;   // M for A, N for B/C
  const int half = lane >> 4;   // selects the K-pair
  const int wid    = (blockIdx.x * blockDim.x + threadIdx.x) >> 5;
  const int nwaves = (gridDim.x * blockDim.x) >> 5;
  const int ntiles = N_EDGES / 16;

  for (int tile = wid; tile < ntiles; tile += nwaves) {
    const int e0 = tile * 16;
    const int er = e0 + row;

    // prefetch next tile's index rows (global_prefetch_b8)
    __builtin_prefetch(src + e0 + nwaves * 16, 0, 0);
    __builtin_prefetch(dst + e0 + nwaves * 16, 0, 0);

    const int   sidx = src[er];
    const int   didx = dst[er];
    const float* hs = hv + (long)sidx * D;
    const float* hd = hv + (long)didx * D;

    // A-fragment for the he part: 16x4, one WMMA
    const v2f ae = *(const v2f*)(he + (long)er * EDIM + half * 2);

    // C init = bias broadcast (bias depends only on column N)
    v8f acc[4];
    #pragma unroll
    for (int nb = 0; nb < 4; ++nb) acc[nb] = splat8(bmsg[nb * 16 + row]);

    // K = 64 over src-part (rows 0..63) and dst-part (rows 64..127)
    #pragma unroll 4
    for (int kk = 0; kk < 16; ++kk) {
      const int k0 = kk * 4 + half * 2;
      const v2f as = *(const v2f*)(hs + k0);
      const v2f ad = *(const v2f*)(hd + k0);
      const int kb = (kk * 2 + half) * D;          // (k0>>1)*64
      #pragma unroll
      for (int nb = 0; nb < 4; ++nb) {
        const int c = nb * 16 + row;
        v2f bs = *(const v2f*)&sW[(kb + c) * 2];              // ds_load_b64
        acc[nb] = wmma4(as, bs, acc[nb]);
        v2f bd = *(const v2f*)&sW[(kb + 32 * D + c) * 2];     // dst part
        acc[nb] = wmma4(ad, bd, acc[nb]);
      }
    }
    // he part: K = 4, one WMMA per output tile (rows 128..131)
    #pragma unroll
    for (int nb = 0; nb < 4; ++nb) {
      const int c = nb * 16 + row;
      v2f be = *(const v2f*)&sW[((64 + half) * D + c) * 2];
      acc[nb] = wmma4(ae, be, acc[nb]);
    }

    // scatter-sum: C element (vgpr r, lane) = msg[e0 + r + half*8, nb*16+row]
    #pragma unroll
    for (int r = 0; r < 8; ++r) {
      const int e = e0 + r + half * 8;
      const int d = dst[e];                  // broadcast load (L2)
      float* arow = acc_a + (long)d * D;
      #pragma unroll
      for (int nb = 0; nb < 4; ++nb)
        atomicAdd(arow + nb * 16 + row, acc[nb][r]);
    }
  }
}

// --------------------------------------------------------------------------
// GRU kernel: per 16-node tile (one wave), per 16-column group g:
//   gi = a•W_ih + b_ih, gh = h•W_hh + b_hh  (6 accumulators, 16 k-steps)
//   r = sig(i_r+h_r); z = sig(i_z+h_z); n = tanh(i_n + r*h_n)
//   h' = (1-z)*n + z*h
// W_ih/W_hh staged cooperatively into interleaved LDS (96 KB):
//   sW[((k>>1)*192 + c)*2 + (k&1)]  ->  B frags are ds_load_2addr_b64.
// --------------------------------------------------------------------------
__global__ void __launch_bounds__(256)
gp_gru(const float* __restrict__ amat,   // [N, 64] aggregated messages
       const float* __restrict__ hv_in,  // [N, 64]
       float*       __restrict__ hv_out, // [N, 64] (may alias hv_in)
       const float* __restrict__ Wih,    // [64, 192]
       const float* __restrict__ Whh,    // [64, 192]
       const float* __restrict__ bih,    // [192]
       const float* __restrict__ bhh)    // [192]
{
  __shared__ float sWi[D * H3];   // 48 KB, interleaved
  __shared__ float sWh[D * H3];   // 48 KB, interleaved

  for (int i = threadIdx.x; i < D * H3; i += blockDim.x) {
    int k = i / H3, c = i % H3;
    int pos = (((k >> 1) * H3) + c) * 2 + (k & 1);
    sWi[pos] = Wih[i];
    sWh[pos] = Whh[i];
  }
  __syncthreads();

  const int lane = threadIdx.x & 31;
  const int row  = lane & 15;
  const int half = lane >> 4;
  const int wid    = (blockIdx.x * blockDim.x + threadIdx.x) >> 5;
  const int nwaves = (gridDim.x * blockDim.x) >> 5;
  const int ntiles = N_NODES / 16;

  for (int tile = wid; tile < ntiles; tile += nwaves) {
    const int node0 = tile * 16;

    #pragma unroll
    for (int g = 0; g < 4; ++g) {
      const int col = g * 16 + row;   // feature column 0..63

      v8f air = splat8(bih[col]);
      v8f aiz = splat8(bih[D + col]);
      v8f ain = splat8(bih[2 * D + col]);
      v8f ahr = splat8(bhh[col]);
      v8f ahz = splat8(bhh[D + col]);
      v8f ahn = splat8(bhh[2 * D + col]);

      #pragma unroll 4
      for (int kk = 0; kk < 16; ++kk) {
        const int k0 = kk * 4 + half * 2;
        const v2f av  = *(const v2f*)(amat  + (long)(node0 + row) * D + k0);
        const v2f hvv = *(const v2f*)(hv_in + (long)(node0 + row) * D + k0);

        const int kb = (kk * 2 + half) * H3;   // (k0>>1)*192

        v2f bir = *(const v2f*)&sWi[(kb + col) * 2];
        v2f biz = *(const v2f*)&sWi[(kb + D + col) * 2];
        v2f bin = *(const v2f*)&sWi[(kb + 2 * D + col) * 2];
        v2f bhr = *(const v2f*)&sWh[(kb + col) * 2];
        v2f bhz = *(const v2f*)&sWh[(kb + D + col) * 2];
        v2f bhn = *(const v2f*)&sWh[(kb + 2 * D + col) * 2];

        air = wmma4(av,  bir, air);
        aiz = wmma4(av,  biz, aiz);
        ain = wmma4(av,  bin, ain);
        ahr = wmma4(hvv, bhr, ahr);
        ahz = wmma4(hvv, bhz, ahz);
        ahn = wmma4(hvv, bhn, ahn);
      }

      // gate math; C element (vgpr r, lane) = row node0+r+half*8, column col
      #pragma unroll
      for (int r = 0; r < 8; ++r) {
        const long m = node0 + r + half * 8;
        const float h  = hv_in[m * D + col];
        const float rr = fast_sigmoid(air[r] + ahr[r]);
        const float z  = fast_sigmoid(aiz[r] + ahz[r]);
        const float n  = fast_tanh(ain[r] + rr * ahn[r]);
        hv_out[m * D + col] = (1.0f - z) * n + z * h;
      }
    }
  }
}

// --------------------------------------------------------------------------
extern "C" void kernel_launch(void* const* d_in, const int* in_sizes, int n_in,
                              void* d_out, int out_size, void* d_ws, size_t ws_size,
                              hipStream_t stream) {
  (void)in_sizes; (void)n_in; (void)out_size; (void)ws_size;

  const float* hv0  = (const float*)d_in[0];   // [N, 64]
  const float* he   = (const float*)d_in[1];   // [E, 4]
  const float* Wmsg = (const float*)d_in[2];   // [R, 132, 64]
  const float* bmsg = (const float*)d_in[3];   // [R, 64]
  const float* Wih  = (const float*)d_in[4];   // [R, 64, 192]
  const float* Whh  = (const float*)d_in[5];   // [R, 64, 192]
  const float* bih  = (const float*)d_in[6];   // [R, 192]
  const float* bhh  = (const float*)d_in[7];   // [R, 192]
  const int*   src  = (const int*)d_in[8];     // [E]
  const int*   dst  = (const int*)d_in[9];     // [E]

  float* hv_out = (float*)d_out;               // [N, 64]
  float* acc_a  = (float*)d_ws;                // [N, 64] scatter accumulator

  const int nA = N_NODES * D;

  for (int t = 0; t < ROUNDS; ++t) {
    const float* hv_cur = (t == 0) ? hv0 : hv_out;

    gp_fill0<<<512, 256, 0, stream>>>(acc_a, nA);

    gp_edge_msg<<<320, 256, 0, stream>>>(
        hv_cur, he,
        Wmsg + (long)t * IN_DIM * D,
        bmsg + (long)t * D,
        src, dst, acc_a);

    gp_gru<<<391, 256, 0, stream>>>(
        acc_a, hv_cur, hv_out,
        Wih + (long)t * D * H3,
        Whh + (long)t * D * H3,
        bih + (long)t * H3,
        bhh + (long)t * H3);
  }
}